// AWGN_modelfree_72198400246424
// MI455X (gfx1250) — compile-verified
//
#include <hip/hip_runtime.h>
#include <hip/hip_bf16.h>

// AWGN channel: out = x + STD * noise, elementwise fp32.
// Memory-bound: 805 MB traffic/call, ~34.5 us floor at 23.3 TB/s.
// Strategy: b128 non-temporal streaming loads/stores, 4 float4 per thread,
// contiguous 16KB-per-stream tiles per block, gfx1250 global_prefetch ahead.

typedef float v4f __attribute__((ext_vector_type(4)));

#define STD_AWGN 0.31622776601683794f  // sqrt(1/10)

// 256 threads * 4 float4 = 1024 float4s (4096 floats) per block tile.
#define T4_PER_BLOCK 1024

__global__ __launch_bounds__(256) void awgn_stream_kernel(
    const v4f* __restrict__ x, const v4f* __restrict__ nz,
    v4f* __restrict__ out, int n4) {
  const int tile = (int)blockIdx.x * T4_PER_BLOCK;
  const int t    = (int)threadIdx.x;

  const int i0 = tile + t;
  const int i1 = i0 + 256;
  const int i2 = i0 + 512;
  const int i3 = i0 + 768;

  // Prefetch the tile 4 blocks ahead into GL2 (gfx1250 global_prefetch_b8).
  // Lanes spread 64B apart to cover the whole 16KB tile per stream.
  {
    const int pf = tile + 4 * T4_PER_BLOCK + t * 4;
    if (pf < n4) {
      __builtin_prefetch((const void*)(x  + pf), 0, 1);
      __builtin_prefetch((const void*)(nz + pf), 0, 1);
    }
  }

  if (i3 < n4) {
    // Fast path (always taken for the 16384x4096 problem): issue all 8
    // b128 NT loads back-to-back for maximum memory-level parallelism.
    v4f x0 = __builtin_nontemporal_load(x  + i0);
    v4f x1 = __builtin_nontemporal_load(x  + i1);
    v4f x2 = __builtin_nontemporal_load(x  + i2);
    v4f x3 = __builtin_nontemporal_load(x  + i3);
    v4f n0 = __builtin_nontemporal_load(nz + i0);
    v4f n1 = __builtin_nontemporal_load(nz + i1);
    v4f n2 = __builtin_nontemporal_load(nz + i2);
    v4f n3 = __builtin_nontemporal_load(nz + i3);

    v4f y0 = x0 + STD_AWGN * n0;   // contracts to v_fmac_f32 per component
    v4f y1 = x1 + STD_AWGN * n1;
    v4f y2 = x2 + STD_AWGN * n2;
    v4f y3 = x3 + STD_AWGN * n3;

    __builtin_nontemporal_store(y0, out + i0);
    __builtin_nontemporal_store(y1, out + i1);
    __builtin_nontemporal_store(y2, out + i2);
    __builtin_nontemporal_store(y3, out + i3);
  } else {
    // Guarded tail (not hit for exact-fit sizes, kept for robustness).
    if (i0 < n4) {
      v4f a = __builtin_nontemporal_load(x + i0);
      v4f b = __builtin_nontemporal_load(nz + i0);
      __builtin_nontemporal_store(a + STD_AWGN * b, out + i0);
    }
    if (i1 < n4) {
      v4f a = __builtin_nontemporal_load(x + i1);
      v4f b = __builtin_nontemporal_load(nz + i1);
      __builtin_nontemporal_store(a + STD_AWGN * b, out + i1);
    }
    if (i2 < n4) {
      v4f a = __builtin_nontemporal_load(x + i2);
      v4f b = __builtin_nontemporal_load(nz + i2);
      __builtin_nontemporal_store(a + STD_AWGN * b, out + i2);
    }
  }
}

extern "C" void kernel_launch(void* const* d_in, const int* in_sizes, int n_in,
                              void* d_out, int out_size, void* d_ws, size_t ws_size,
                              hipStream_t stream) {
  const float* x  = (const float*)d_in[0];
  const float* nz = (const float*)d_in[1];
  float* out      = (float*)d_out;

  // 16384 * 4096 = 67,108,864 elements; divisible by 4 (row length 4096).
  const long long n  = (long long)in_sizes[0];
  const int       n4 = (int)(n >> 2);

  const int blocks = (n4 + T4_PER_BLOCK - 1) / T4_PER_BLOCK;  // 16384
  awgn_stream_kernel<<<blocks, 256, 0, stream>>>(
      (const v4f*)x, (const v4f*)nz, (v4f*)out, n4);
}